// CausalDepthwiseMixer_21869973471733
// MI455X (gfx1250) — compile-verified
//
#include <hip/hip_runtime.h>
#include <stdint.h>

// Problem constants (match reference).
#define BATCH   4
#define SEQ     4096
#define CDIM    2048
#define KSZ     7

// Tiling.
#define THREADS 256            // 8 waves (wave32)
#define CB      1024           // channels per block = THREADS * 4
#define TC      8              // time rows per TDM chunk  (32 KB / chunk)
#define TSB     64             // time rows per block strip
#define NCHUNK  (TSB / TC)     // 8 chunks, double-buffered

typedef uint32_t v4u  __attribute__((ext_vector_type(4)));
typedef int      v4i  __attribute__((ext_vector_type(4)));
typedef int      v8i  __attribute__((ext_vector_type(8)));
typedef float    f32x4 __attribute__((ext_vector_type(4)));

// ---- CDNA5 wait: TENSORcnt --------------------------------------------------
#if __has_builtin(__builtin_amdgcn_s_wait_tensorcnt)
#define WAIT_TENSORCNT(N) __builtin_amdgcn_s_wait_tensorcnt(N)
#else
#define WAIT_TENSORCNT(N) asm volatile("s_wait_tensorcnt %0" ::"i"(N) : "memory")
#endif

#define HAVE_TDM __has_builtin(__builtin_amdgcn_tensor_load_to_lds)

// ---- TDM: DMA one [TC x CB] fp32 tile (row stride CDIM) into LDS -----------
// D# packed per CDNA5 ISA ch.8. 2D tensor -> groups 2/3 zero.
__device__ __forceinline__ void tdm_load_tile(uint32_t lds_byte_off,
                                              const float* gsrc) {
  uint64_t ga = (uint64_t)(uintptr_t)gsrc;

  v4u g0;
  g0[0] = 0x1u;                                        // count=1, user mode
  g0[1] = lds_byte_off;                                // lds_addr [63:32]
  g0[2] = (uint32_t)ga;                                // global_addr lo
  g0[3] = ((uint32_t)(ga >> 32) & 0x01FFFFFFu)         // global_addr[56:32]
          | (2u << 30);                                // type=2 ("image")

  v8i g1;
  g1[0] = (int)(2u << 16);                             // data_size=4B; mask=0
  g1[1] = (int)(((uint32_t)CB & 0xFFFFu) << 16);       // tensor_dim0 lo16
  g1[2] = (int)((((uint32_t)CB >> 16) & 0xFFFFu)       // tensor_dim0 hi16
                | ((uint32_t)TC << 16));               // tensor_dim1 lo16
  g1[3] = (int)(((uint32_t)CB & 0xFFFFu) << 16);       // dim1 hi=0 | tile_dim0
  g1[4] = (int)TC;                                     // tile_dim1 | tile_dim2=0
  g1[5] = (int)CDIM;                                   // tensor_dim0_stride lo32
  g1[6] = 0;                                           // stride0 hi | stride1 lo
  g1[7] = 0;                                           // stride1 hi

  v4i z4 = {0, 0, 0, 0};
#if HAVE_TDM
#if __has_include(<hip/amd_detail/amd_gfx1250_TDM.h>)
  // amdgpu-toolchain (clang-23): 6-arg form
  v8i z8 = {0, 0, 0, 0, 0, 0, 0, 0};
  __builtin_amdgcn_tensor_load_to_lds(g0, g1, z4, z4, z8, 0);
#else
  // ROCm 7.2 (clang-22): 5-arg form
  __builtin_amdgcn_tensor_load_to_lds(g0, g1, z4, z4, 0);
#endif
#else
  (void)g0; (void)g1; (void)z4;   // fallback copy done in-kernel
#endif
}

__device__ __forceinline__ f32x4 fma4(f32x4 a, f32x4 b, f32x4 c) {
  f32x4 r;
  r.x = __builtin_fmaf(a.x, b.x, c.x);
  r.y = __builtin_fmaf(a.y, b.y, c.y);
  r.z = __builtin_fmaf(a.z, b.z, c.z);
  r.w = __builtin_fmaf(a.w, b.w, c.w);
  return r;
}

// ---- Kernel ----------------------------------------------------------------
// out[b,t,c] = bias[c] + sum_{k=0..6} w[c,0,k] * x[b, t-6+k, c]   (zero-pad t<0)
__global__ __launch_bounds__(THREADS)
void CausalDepthwiseMixer_kernel(const float* __restrict__ xg,
                                 const float* __restrict__ wg,
                                 const float* __restrict__ bg,
                                 float* __restrict__ outg) {
  __shared__ float smem[2][TC][CB];   // 64 KB double buffer

  const int tid = threadIdx.x;
  const int cg  = blockIdx.x;         // channel group (0..1)
  const int tg  = blockIdx.y;         // time strip (0..63)
  const int bb  = blockIdx.z;         // batch

  const int c0g = cg * CB + tid * 4;  // this lane's first channel
  const int t0  = tg * TSB;

  const float* xb   = xg + (size_t)bb * SEQ * CDIM;
  const float* tile = xb + (size_t)t0 * CDIM + (size_t)cg * CB; // TDM origin
  float*       ob   = outg + (size_t)bb * SEQ * CDIM;

  // LDS byte offsets of the two buffers. Generic shared pointers carry the
  // LDS byte offset in their low 32 bits (flat LDS aperture: addr[31:0]).
  const uint32_t lds0 = (uint32_t)(uintptr_t)(void*)&smem[0][0][0];
  const uint32_t lds1 = (uint32_t)(uintptr_t)(void*)&smem[1][0][0];

  // Wave 0 is the sole TDM issuer/waiter (TENSORcnt is per-wave; TDM ignores
  // EXEC, so other waves must *branch around* the issue — verified to compile
  // to s_and_saveexec + s_cbranch_execz).
  const bool issuer = (tid < 32);

  // Kick chunk 0 immediately to overlap with weight/halo setup.
  if (issuer) tdm_load_tile(lds0, tile);

  // Per-lane weights (4 channels x 7 taps) — torch layout w[c][0][k].
  f32x4 wk[KSZ];
#pragma unroll
  for (int k = 0; k < KSZ; ++k) {
    wk[k].x = wg[(size_t)(c0g + 0) * KSZ + k];
    wk[k].y = wg[(size_t)(c0g + 1) * KSZ + k];
    wk[k].z = wg[(size_t)(c0g + 2) * KSZ + k];
    wk[k].w = wg[(size_t)(c0g + 3) * KSZ + k];
  }
  const f32x4 bias = *(const f32x4*)(bg + c0g);

  // Sliding-window history: x[t0-6 .. t0-1] (zero before sequence start).
  f32x4 h[KSZ - 1];
#pragma unroll
  for (int j = 0; j < KSZ - 1; ++j) {
    const int t = t0 - (KSZ - 1) + j;
    if (t >= 0) {
      h[j] = *(const f32x4*)(xb + (size_t)t * CDIM + c0g);
    } else {
      h[j] = (f32x4){0.f, 0.f, 0.f, 0.f};
    }
  }

  for (int ci = 0; ci < NCHUNK; ++ci) {
#if !HAVE_TDM
    // Fallback (no TDM builtin): cooperative copy of chunk ci.
    {
      const float* src = tile + (size_t)ci * TC * CDIM;
      float* dst = &smem[ci & 1][0][0];
#pragma unroll
      for (int r = 0; r < TC; ++r)
        *(f32x4*)(dst + r * CB + tid * 4) =
            *(const f32x4*)(src + (size_t)r * CDIM + tid * 4);
    }
#endif
    if (issuer) {
      if (ci + 1 < NCHUNK) {
        // Prefetch next chunk into the other buffer, then wait until only it
        // is outstanding (TDM completes in order within a wave).
        tdm_load_tile((ci & 1) ? lds0 : lds1, tile + (size_t)(ci + 1) * TC * CDIM);
        WAIT_TENSORCNT(1);
      } else {
        WAIT_TENSORCNT(0);
      }
    }
    __syncthreads();                     // buf[ci&1] now valid for all waves
    asm volatile("" ::: "memory");       // keep LDS loads below the wait

    const float* lrow = &smem[ci & 1][0][tid * 4];
#pragma unroll
    for (int r = 0; r < TC; ++r) {
      const f32x4 cur = *(const f32x4*)(lrow + r * CB);   // ds_load_b128
      f32x4 acc = bias;
#pragma unroll
      for (int k = 0; k < KSZ - 1; ++k) acc = fma4(wk[k], h[k], acc);
      acc = fma4(wk[KSZ - 1], cur, acc);
#pragma unroll
      for (int k = 0; k < KSZ - 2; ++k) h[k] = h[k + 1];
      h[KSZ - 2] = cur;

      const int t = t0 + ci * TC + r;
      // Streaming output: never re-read -> non-temporal b128 store, keep L2
      // for the x halo rows that neighboring strips re-read.
      __builtin_nontemporal_store(acc, (f32x4*)(ob + (size_t)t * CDIM + c0g));
    }
    __syncthreads();  // all readers done before buf[ci&1] is re-filled (ci+2)
  }
}

// ---- Host launcher ----------------------------------------------------------
extern "C" void kernel_launch(void* const* d_in, const int* in_sizes, int n_in,
                              void* d_out, int out_size, void* d_ws, size_t ws_size,
                              hipStream_t stream) {
  (void)in_sizes; (void)n_in; (void)out_size; (void)d_ws; (void)ws_size;
  const float* x = (const float*)d_in[0];   // [B, T, C] fp32
  const float* w = (const float*)d_in[1];   // [C, 1, K] fp32
  const float* b = (const float*)d_in[2];   // [C]       fp32
  float* out = (float*)d_out;               // [B, T, C] fp32

  dim3 grid(CDIM / CB, SEQ / TSB, BATCH);   // (2, 64, 4) = 512 blocks
  dim3 block(THREADS);
  CausalDepthwiseMixer_kernel<<<grid, block, 0, stream>>>(x, w, b, out);
}